// LiGO_214748365063
// MI455X (gfx1250) — compile-verified
//
#include <hip/hip_runtime.h>

typedef float v2f __attribute__((ext_vector_type(2)));
typedef float v8f __attribute__((ext_vector_type(8)));
typedef float f4  __attribute__((ext_vector_type(4)));

#define NL1   6
#define NL2   12
#define H1D   1024
#define H2D   2048
#define NB    4096

#define BM 128
#define BN 128
#define BK 32
#define LDSS (BK + 4)        // 36-float row stride (144 B, 16B-aligned chunks, no bank conflicts)
#define TILE (BM * LDSS)     // floats per staged tile

// ---- CDNA5 async global->LDS copy (GLOBAL_LOAD_ASYNC_TO_LDS_B128, ASYNCcnt) ----
__device__ __forceinline__ void async_copy16(const float* g, unsigned lds_byte_off)
{
    asm volatile("global_load_async_to_lds_b128 %0, %1, off"
                 :: "v"(lds_byte_off), "v"(g)
                 : "memory");
}
__device__ __forceinline__ void wait_async0()
{
    asm volatile("s_wait_asynccnt 0x0" ::: "memory");
}

// ---------------------------------------------------------------------------
// Generic NT GEMM: C[m,n] = sum_k A[m,k]*B[n,k]  (+ bias[n], optional ReLU)
// A: M x K row-major, B: N x K row-major, C: M x N row-major.
// M % 128 == 0, N % 128 == 0, K % 32 == 0 (true for all uses here).
// blockIdx.z selects a batched slice via strideA/B/C.
// V_WMMA_F32_16X16X4_F32 main loop, double-buffered async LDS staging.
// ---------------------------------------------------------------------------
template<bool BIAS, bool RELU>
__global__ __launch_bounds__(256)
void gemm_nt_wmma(const float* __restrict__ A, const float* __restrict__ B,
                  const float* __restrict__ bias, float* __restrict__ C,
                  int M, int N, int K,
                  long strideA, long strideB, long strideC)
{
    __shared__ float smem[2 * 2 * TILE];   // [buf][A|B][BM*LDSS]

    const int z = blockIdx.z;
    A += (long)z * strideA;
    B += (long)z * strideB;
    C += (long)z * strideC;

    const int tid    = threadIdx.x;
    const int wave   = tid >> 5;        // 8 waves of 32 (wave32)
    const int lane   = tid & 31;
    const int wm     = wave >> 2;       // 0..1  -> 64-row strip
    const int wn     = wave & 3;        // 0..3  -> 32-col strip
    const int m_blk  = blockIdx.y * BM;
    const int n_blk  = blockIdx.x * BN;
    const int lane15 = lane & 15;
    const int khalf  = lane >> 4;       // 16-lane half selects K pair

    const unsigned smem_base = (unsigned)(unsigned long long)(uintptr_t)smem;

    // issue async copies of one 128x32 A tile + one 128x32 B tile into buffer `buf`
    auto issue_tiles = [&](int kt, int buf) {
        unsigned aoff = smem_base + (unsigned)(buf * 2 * TILE) * 4u;
        unsigned boff = aoff + (unsigned)TILE * 4u;
#pragma unroll
        for (int p = 0; p < 4; ++p) {
            int f   = tid + p * 256;        // float4 chunk index 0..1023
            int row = f >> 3;               // 8 chunks per 32-float row
            int col = (f & 7) << 2;
            unsigned loff = (unsigned)(row * LDSS + col) * 4u;
            async_copy16(A + (long)(m_blk + row) * K + kt + col, aoff + loff);
            async_copy16(B + (long)(n_blk + row) * K + kt + col, boff + loff);
        }
    };

    v8f acc[4][2];
#pragma unroll
    for (int tm = 0; tm < 4; ++tm)
#pragma unroll
        for (int tn = 0; tn < 2; ++tn)
            acc[tm][tn] = (v8f){0.f,0.f,0.f,0.f,0.f,0.f,0.f,0.f};

    // prologue: fill buffer 0
    issue_tiles(0, 0);
    wait_async0();
    __syncthreads();

    int cur = 0;
    for (int kt = 0; kt < K; kt += BK) {
        // prefetch next K-tile into the other buffer while we compute
        if (kt + BK < K) issue_tiles(kt + BK, cur ^ 1);

        const float* Asc = smem + cur * 2 * TILE;
        const float* Bsc = Asc + TILE;

#pragma unroll
        for (int kk = 0; kk < BK; kk += 4) {
            v2f a[4], b[2];
#pragma unroll
            for (int tm = 0; tm < 4; ++tm)
                a[tm] = *(const v2f*)&Asc[(wm * 64 + tm * 16 + lane15) * LDSS + kk + 2 * khalf];
#pragma unroll
            for (int tn = 0; tn < 2; ++tn)
                b[tn] = *(const v2f*)&Bsc[(wn * 32 + tn * 16 + lane15) * LDSS + kk + 2 * khalf];
#pragma unroll
            for (int tm = 0; tm < 4; ++tm)
#pragma unroll
                for (int tn = 0; tn < 2; ++tn)
                    acc[tm][tn] = __builtin_amdgcn_wmma_f32_16x16x4_f32(
                        false, a[tm], false, b[tn], (short)0, acc[tm][tn], false, false);
        }

        wait_async0();       // next buffer's data landed in LDS
        __syncthreads();     // everyone done reading cur, writes visible
        cur ^= 1;
    }

    // ---- epilogue: C/D layout -> VGPR r holds rows r (lanes 0-15), r+8 (16-31)
#pragma unroll
    for (int tm = 0; tm < 4; ++tm) {
        int row0 = m_blk + wm * 64 + tm * 16 + 8 * khalf;
#pragma unroll
        for (int tn = 0; tn < 2; ++tn) {
            int col = n_blk + wn * 32 + tn * 16 + lane15;
            float bv = BIAS ? bias[col] : 0.0f;
#pragma unroll
            for (int r = 0; r < 8; ++r) {
                float v = acc[tm][tn][r] + bv;
                if (RELU) v = fmaxf(v, 0.0f);
                C[(long)(row0 + r) * N + col] = v;
            }
        }
    }
}

// B1e[l][o] = sum_h b_b[l][o][h] * B1[l][h]   (one wave32 per output)
__global__ __launch_bounds__(256)
void bias_expand_kernel(const float* __restrict__ b_b, const float* __restrict__ B1,
                        float* __restrict__ B1e)
{
    int wgid = blockIdx.x * 8 + (threadIdx.x >> 5);
    int lane = threadIdx.x & 31;
    int l = wgid >> 11;           // / 2048
    int o = wgid & (H2D - 1);
    const float* row = b_b + ((long)l * H2D + o) * H1D;
    const float* bv  = B1  + (long)l * H1D;
    float s = 0.f;
    for (int h = lane; h < H1D; h += 32) s += row[h] * bv[h];
#pragma unroll
    for (int off = 16; off > 0; off >>= 1) s += __shfl_xor(s, off, 32);
    if (lane == 0) B1e[(long)l * H2D + o] = s;
}

// W2[j] = sum_i w[j,i] * W1e[i]   (float4 per thread, 12 outputs per read-set)
__global__ __launch_bounds__(256)
void w2_combine_kernel(const float* __restrict__ w, const float* __restrict__ W1e,
                       float* __restrict__ W2out)
{
    const long NE4 = (long)H2D * H2D / 4;
    long p = (long)blockIdx.x * 256 + threadIdx.x;
    f4 v[NL1];
#pragma unroll
    for (int i = 0; i < NL1; ++i) v[i] = ((const f4*)W1e)[(long)i * NE4 + p];
#pragma unroll
    for (int j = 0; j < NL2; ++j) {
        f4 a = (f4){0.f, 0.f, 0.f, 0.f};
#pragma unroll
        for (int i = 0; i < NL1; ++i) { float c = w[j * NL1 + i]; a = a + v[i] * c; }
        ((f4*)W2out)[(long)j * NE4 + p] = a;
    }
}

// B2[j][o] = sum_i w[j,i] * B1e[i][o]
__global__ __launch_bounds__(256)
void b2_kernel(const float* __restrict__ w, const float* __restrict__ B1e,
               float* __restrict__ B2out)
{
    int idx = blockIdx.x * blockDim.x + threadIdx.x;
    if (idx >= NL2 * H2D) return;
    int j = idx >> 11;
    int o = idx & (H2D - 1);
    float s = 0.f;
#pragma unroll
    for (int i = 0; i < NL1; ++i) s += w[j * NL1 + i] * B1e[i * H2D + o];
    B2out[idx] = s;
}

extern "C" void kernel_launch(void* const* d_in, const int* in_sizes, int n_in,
                              void* d_out, int out_size, void* d_ws, size_t ws_size,
                              hipStream_t stream)
{
    const float* x   = (const float*)d_in[0];
    const float* W1  = (const float*)d_in[1];
    const float* B1  = (const float*)d_in[2];
    // d_in[3] = W20, d_in[4] = B20: shape templates only, values unused
    const float* w   = (const float*)d_in[5];
    const float* a_w = (const float*)d_in[6];
    const float* b_w = (const float*)d_in[7];
    const float* b_b = (const float*)d_in[8];

    float* out_h  = (float*)d_out;                       // [4096, 2048]
    float* out_W2 = out_h + (long)NB * H2D;              // [12, 2048, 2048]
    float* out_B2 = out_W2 + (long)NL2 * H2D * H2D;      // [12, 2048]

    float* ws  = (float*)d_ws;
    float* W1e = ws;                                     // 6*2048*2048
    float* M1t = W1e + (long)NL1 * H2D * H2D;            // 6*2048*1024
    float* B1e = M1t + (long)NL1 * H2D * H1D;            // 6*2048
    // h ping-pong reuses the W1e region (free once w2_combine has run)
    float* hA  = ws;
    float* hB  = hA + (long)NB * H2D;

    dim3 blk(256);

    // 1) M1t[l] = a_w[l] @ W1[l]^T        (M=2048, N=1024, K=1024, z=6)
    gemm_nt_wmma<false, false><<<dim3(H1D / BN, H2D / BM, NL1), blk, 0, stream>>>(
        a_w, W1, nullptr, M1t, H2D, H1D, H1D,
        (long)H2D * H1D, (long)H1D * H1D, (long)H2D * H1D);

    // 2) W1e[l] = b_w[l] @ M1t[l]^T       (M=2048, N=2048, K=1024, z=6)
    gemm_nt_wmma<false, false><<<dim3(H2D / BN, H2D / BM, NL1), blk, 0, stream>>>(
        b_w, M1t, nullptr, W1e, H2D, H2D, H1D,
        (long)H2D * H1D, (long)H2D * H1D, (long)H2D * H2D);

    // 3) B1e = einsum('loh,lh->lo')
    bias_expand_kernel<<<(NL1 * H2D) / 8, blk, 0, stream>>>(b_b, B1, B1e);

    // 4) W2 = w x W1e  -> d_out ;  B2 = w x B1e -> d_out
    w2_combine_kernel<<<(unsigned)((long)H2D * H2D / 4 / 256), blk, 0, stream>>>(w, W1e, out_W2);
    b2_kernel<<<(NL2 * H2D + 255) / 256, blk, 0, stream>>>(w, B1e, out_B2);

    // 5) 12-layer MLP: h = relu(h @ W2[l]^T + B2[l]); last layer no ReLU
    const float* cur = x;
    float* nxt = hA;
    for (int l = 0; l < NL2; ++l) {
        bool last = (l == NL2 - 1);
        float* dst = last ? out_h : nxt;
        const float* Bm = out_W2 + (long)l * H2D * H2D;
        const float* bs = out_B2 + (long)l * H2D;
        if (!last)
            gemm_nt_wmma<true, true><<<dim3(H2D / BN, NB / BM, 1), blk, 0, stream>>>(
                cur, Bm, bs, dst, NB, H2D, H2D, 0, 0, 0);
        else
            gemm_nt_wmma<true, false><<<dim3(H2D / BN, NB / BM, 1), blk, 0, stream>>>(
                cur, Bm, bs, dst, NB, H2D, H2D, 0, 0, 0);
        cur = dst;
        nxt = (nxt == hA) ? hB : hA;
    }
}